// LocalEmbeddingLayer_48550310314553
// MI455X (gfx1250) — compile-verified
//
#include <hip/hip_runtime.h>
#include <hip/hip_bf16.h>
#include <math.h>

// Shapes from the reference
#define BB   32
#define NN   1024
#define FF   64      // feature dim
#define PP   128     // output dim
#define KK   16      // neighbors kept
#define H2   256     // 2*P hidden
#define F2   128     // 2*F

typedef __attribute__((ext_vector_type(2))) float v2f;
typedef __attribute__((ext_vector_type(4))) float v4f;
typedef __attribute__((ext_vector_type(8))) float v8f;
typedef __attribute__((ext_vector_type(4))) int   i128_t;   // 16B payload type

#if defined(__gfx1250__) && __has_builtin(__builtin_amdgcn_global_load_async_to_lds_b128)
#define HAVE_ASYNC 1
#else
#define HAVE_ASYNC 0
#endif

// Builtin expects pointers to 4xi32 vectors: global src (AS1), LDS dst (AS3).
#define AS1P(p) ((__attribute__((address_space(1))) i128_t*)(p))
#define AS3P(p) ((__attribute__((address_space(3))) i128_t*)(p))

__device__ __forceinline__ void wait_async0() {
#if __has_builtin(__builtin_amdgcn_s_wait_asynccnt)
    __builtin_amdgcn_s_wait_asynccnt(0);
    asm volatile("" ::: "memory");
#else
    asm volatile("s_wait_asynccnt 0x0" ::: "memory");
#endif
}

__device__ __forceinline__ float gelu_exact(float x) {
    return 0.5f * x * (1.0f + erff(x * 0.70710678118654752440f));
}

// ---------------------------------------------------------------------------
// Kernel 0: prep weights into workspace.
//  W1' = [W1_top ; W1_bot - W1_top]  (folds the (nb - c) subtraction into W1)
//  Both W1' and W2 stored row-PAIR interleaved:  w[(r>>1)*2C + c*2 + (r&1)]
//  so a lane's (r, r+1) B-operand pair is one 8-byte ds_load_b64.
// ---------------------------------------------------------------------------
__global__ void __launch_bounds__(256)
prep_kernel(const float* __restrict__ W1, const float* __restrict__ W2,
            float* __restrict__ W1i, float* __restrict__ W2i) {
    int t = blockIdx.x * 256 + threadIdx.x;      // 0 .. 32767
    {   // W1: 128 x 256
        int r = t >> 8, c = t & 255;
        float v = W1[r * H2 + c];
        if (r >= FF) v -= W1[(r - FF) * H2 + c];
        W1i[(r >> 1) * (2 * H2) + c * 2 + (r & 1)] = v;
    }
    {   // W2: 256 x 128
        int r = t >> 7, c = t & 127;
        W2i[(r >> 1) * (2 * PP) + c * 2 + (r & 1)] = W2[r * PP + c];
    }
}

// ---------------------------------------------------------------------------
// Kernel 1: exact KNN (top-17 smallest squared distances, drop self at rank 0)
// ---------------------------------------------------------------------------
__global__ void __launch_bounds__(256)
knn_kernel(const float* __restrict__ points, int* __restrict__ knn_idx) {
    __shared__ float sx[NN], sy[NN], sz[NN], sr[NN];
    const int b   = blockIdx.y;
    const int tid = threadIdx.x;
    const float* pb = points + (size_t)b * NN * 3;

    for (int t = tid; t < NN; t += 256) {
        float x = pb[t * 3 + 0];
        float y = pb[t * 3 + 1];
        float z = pb[t * 3 + 2];
        sx[t] = x; sy[t] = y; sz[t] = z;
        sr[t] = x * x + y * y + z * z;
    }
    __syncthreads();

    const int n = blockIdx.x * 256 + tid;
    const float xn = sx[n], yn = sy[n], zn = sz[n], rn = sr[n];

    float dist[KK + 1];
    int   ind [KK + 1];
#pragma unroll
    for (int j = 0; j <= KK; ++j) { dist[j] = 3.4e38f; ind[j] = 0; }

    for (int m = 0; m < NN; ++m) {
        float d = rn + sr[m] - 2.0f * (xn * sx[m] + yn * sy[m] + zn * sz[m]);
        if (d < dist[KK]) {
            float dj = d; int ij = m;
#pragma unroll
            for (int j = 0; j <= KK; ++j) {     // stable sorted insert (strict <)
                bool sw = dj < dist[j];
                float td = dist[j]; int ti = ind[j];
                dist[j] = sw ? dj : td;  ind[j] = sw ? ij : ti;
                dj      = sw ? td : dj;  ij     = sw ? ti : ij;
            }
        }
    }
    int* o = knn_idx + ((size_t)b * NN + n) * KK;
#pragma unroll
    for (int j = 0; j < KK; ++j) o[j] = ind[j + 1];   // drop self (rank 0)
}

// ---------------------------------------------------------------------------
// Kernel 2: gather + 2-layer GELU MLP + mean over K on v_wmma_f32_16x16x4_f32.
// One wave per (b,n) job; A-matrix is a raw [nb | c] copy (async-to-LDS).
//
// Dynamic LDS (floats): W1i 32768 | W2i 32768 | per-wave 2048 A + 256 H
// Total 292 KB < 320 KB WGP LDS.
// ---------------------------------------------------------------------------
#define WAVES_PER_WG 4
#define WG_THREADS   (WAVES_PER_WG * 32)
#define LDS_FLOATS   (65536 + WAVES_PER_WG * (2048 + 256))

__global__ void __launch_bounds__(WG_THREADS)
mlp_kernel(const float* __restrict__ features,
           const int*   __restrict__ knn_idx,
           const float* __restrict__ wts,      // W1i (32768) then W2i (32768)
           const float* __restrict__ b1, const float* __restrict__ b2,
           float* __restrict__ out) {
    extern __shared__ float smem[];
    float* sW1 = smem;                 // pair-interleaved W1' (128x256)
    float* sW2 = smem + 32768;         // pair-interleaved W2  (256x128)

    const int wave = threadIdx.x >> 5;
    const int lane = threadIdx.x & 31;
    float* sA = smem + 65536 + wave * (2048 + 256);   // 16x128 [nb | c]
    float* sH = sA + 2048;                            // 16x16 H-tile staging

    // ---- Stage 256KB of interleaved weights into LDS (linear copy) ----
#if HAVE_ASYNC
    {
        const float* src = wts  + threadIdx.x * 4;
        float*       dst = smem + threadIdx.x * 4;
        for (int it = 0; it < 65536 / (WG_THREADS * 4); ++it) {
            __builtin_amdgcn_global_load_async_to_lds_b128(
                AS1P(src), AS3P(dst), 0, 0);
            src += WG_THREADS * 4;
            dst += WG_THREADS * 4;
        }
        wait_async0();
    }
#else
    for (int t = threadIdx.x * 4; t < 65536; t += WG_THREADS * 4)
        *(v4f*)(smem + t) = *(const v4f*)(wts + t);
#endif
    __syncthreads();

    const int mrow = lane & 15;                 // M row / N col this lane owns
    const int cadd = (lane < 16) ? 0 : 2;       // K sub-offset (dual layout)
    const int par  = (lane < 16) ? 0 : 1;       // which (r,r+1) pair of the 4-row K step
    const int half = lane >> 4;                 // gather: 0=neighbor, 1=center
    const int q16  = (lane & 15) * 4;           // 16B chunk within a 64-float row

    // Job-invariant per-lane biases.
    float bias1r[16], bias2r[8];
#pragma unroll
    for (int j = 0; j < 16; ++j) bias1r[j] = b1[j * 16 + mrow];
#pragma unroll
    for (int t = 0; t < 8; ++t)  bias2r[t] = b2[t * 16 + mrow];

    const int nWaves = gridDim.x * WAVES_PER_WG;
    for (int job = blockIdx.x * WAVES_PER_WG + wave; job < BB * NN; job += nWaves) {
        const int b = job >> 10;
        const int n = job & (NN - 1);
        const float* fb = features + (size_t)b * NN * FF;
        const int*   nb = knn_idx + (size_t)job * KK;

        // ---- Build A = [neighbor | center] (16 x 128) via pure copy ----
        asm volatile("s_wait_dscnt 0x0" ::: "memory");  // prior job's reads done
#if HAVE_ASYNC
#pragma unroll
        for (int i = 0; i < 16; ++i) {          // row i: lanes0-15 nb, 16-31 c
            int row = half ? n : nb[i];
            const float* src = fb + row * FF + q16;
            float*       dst = sA + i * F2 + half * FF + q16;
            __builtin_amdgcn_global_load_async_to_lds_b128(
                AS1P(src), AS3P(dst), 0, 0);
        }
        wait_async0();
#else
#pragma unroll
        for (int i = 0; i < 16; ++i) {
            int row = half ? n : nb[i];
            *(v4f*)(sA + i * F2 + half * FF + q16) =
                *(const v4f*)(fb + row * FF + q16);
        }
        asm volatile("s_wait_dscnt 0x0" ::: "memory");
#endif

        // ---- Load A operand registers (one ds_load_b64 each) ----
        v2f a[32];
#pragma unroll
        for (int kk = 0; kk < 32; ++kk)
            a[kk] = *(const v2f*)(sA + mrow * F2 + kk * 4 + cadd);

        v8f acc[8];
#pragma unroll
        for (int t = 0; t < 8; ++t) acc[t] = (v8f){0.f,0.f,0.f,0.f,0.f,0.f,0.f,0.f};

        for (int j = 0; j < 16; ++j) {          // hidden N-tiles (16x16)
            // ---- GEMM1: H_j = [nb|c] @ W1'[:, 16j:16j+16] ----
            v8f h = (v8f){0.f,0.f,0.f,0.f,0.f,0.f,0.f,0.f};
#pragma unroll
            for (int kk = 0; kk < 32; ++kk) {
                // pair index = 2kk + par ; one b64 load
                v2f wb = *(const v2f*)(sW1 + (2 * kk + par) * (2 * H2)
                                           + (j * 16 + mrow) * 2);
                h = __builtin_amdgcn_wmma_f32_16x16x4_f32(
                        false, a[kk], false, wb, (short)0, h, false, false);
            }
            float bias1 = bias1r[j];
#pragma unroll
            for (int r = 0; r < 8; ++r) h[r] = gelu_exact(h[r] + bias1);

            // Stage H_j (D-layout) -> LDS -> reload in A-layout.
#pragma unroll
            for (int r = 0; r < 8; ++r) {
                int M = r + ((lane < 16) ? 0 : 8);
                sH[M * 16 + mrow] = h[r];
            }
            asm volatile("s_wait_dscnt 0x0" ::: "memory");

            // ---- GEMM2 partial: acc += gelu(H_j) @ W2[16j:16j+16, :] ----
#pragma unroll
            for (int kk2 = 0; kk2 < 4; ++kk2) {
                int kcol = kk2 * 4 + cadd;
                v2f ha = *(const v2f*)(sH + mrow * 16 + kcol);
                int pair = j * 8 + 2 * kk2 + par;   // (rr>>1) for rr=j*16+kcol
#pragma unroll
                for (int t = 0; t < 8; ++t) {
                    v2f wb2 = *(const v2f*)(sW2 + pair * (2 * PP)
                                                + (t * 16 + mrow) * 2);
                    acc[t] = __builtin_amdgcn_wmma_f32_16x16x4_f32(
                                false, ha, false, wb2, (short)0, acc[t], false, false);
                }
            }
        }

        // ---- Epilogue: bias2 + GELU, mean over K=16 rows, store 128 ----
#pragma unroll
        for (int t = 0; t < 8; ++t) {
            float s = 0.0f;
#pragma unroll
            for (int r = 0; r < 8; ++r) s += gelu_exact(acc[t][r] + bias2r[t]);
            s += __shfl_xor(s, 16, 32);         // combine M=0..7 with M=8..15
            if (lane < 16)
                out[(size_t)job * PP + t * 16 + lane] = s * (1.0f / 16.0f);
        }
    }
}

// ---------------------------------------------------------------------------
extern "C" void kernel_launch(void* const* d_in, const int* in_sizes, int n_in,
                              void* d_out, int out_size, void* d_ws, size_t ws_size,
                              hipStream_t stream) {
    const float* points   = (const float*)d_in[0];
    const float* features = (const float*)d_in[1];
    const float* W1       = (const float*)d_in[2];
    const float* b1       = (const float*)d_in[3];
    const float* W2       = (const float*)d_in[4];
    const float* b2       = (const float*)d_in[5];
    float*       out      = (float*)d_out;

    int*   knn = (int*)d_ws;                             // 2 MB of indices
    float* wts = (float*)((char*)d_ws + (size_t)BB * NN * KK * sizeof(int));
    float* W1i = wts;                                    // 32768 floats
    float* W2i = wts + 32768;                            // 32768 floats

    prep_kernel<<<128, 256, 0, stream>>>(W1, W2, W1i, W2i);

    dim3 g1(NN / 256, BB);
    knn_kernel<<<g1, 256, 0, stream>>>(points, knn);

    const size_t ldsBytes = (size_t)LDS_FLOATS * sizeof(float);   // 292 KB
    (void)hipFuncSetAttribute((const void*)mlp_kernel,
                              hipFuncAttributeMaxDynamicSharedMemorySize,
                              (int)ldsBytes);
    mlp_kernel<<<1024, WG_THREADS, ldsBytes, stream>>>(
        features, knn, wts, b1, b2, out);
}